// SingleHeadNeuralMemoryLinear_26989574488240
// MI455X (gfx1250) — compile-verified
//
#include <hip/hip_runtime.h>
#include <math.h>

// ---------------------------------------------------------------------------
// SingleHeadNeuralMemoryLinear for MI455X (gfx1250, wave32, WMMA + TDM).
//
// All heavy math is fp32 GEMM -> V_WMMA_F32_16X16X4_F32.
// Global->LDS staging of K-contiguous operands uses the Tensor Data Mover
// (tensor_load_to_lds, TENSORcnt) with double-buffered LDS tiles and a
// one-tile software pipeline; TDM's pad feature (2 DWORDs per 32) reproduces
// the bank-conflict-avoiding BK+2 row stride.
//
// Pipeline (all on `stream`):
//   1. rowdot_lrwd : lr[b,l], log_wd[b,l]           (wave-per-row dot)
//   2. scan_decay  : cwd=cumsum(log_wd), wd_cross=exp(cwd),
//                    gneg[l]=-lr[l]*exp(cwd[L-1]-cwd[l]), wdl[b]=exp(cwd[L-1])
//   3. Q = x@w_q+b_q ; 4. K = x@w_k+b_k ; 5. V = x@w_v+b_v
//   6. WpKV = K@Wpᵀ - V                (in place over V)
//   7. QWs  = (Q@Wpᵀ) * wd_cross[row]
//   8. T[j,i] = (j>=i) ? -lr[i]*exp(cwd[j]-cwd[i])*(K@Qᵀ)[i,j] : 0
//   9. y = T@WpKV + QWs               -> d_out[0 .. B*L*D)
//  10. W_next = (gneg⊙WpKV)ᵀ@K + wdl*W_prev -> d_out[B*L*D ..)
// ---------------------------------------------------------------------------

typedef __attribute__((ext_vector_type(2))) float v2f;
typedef __attribute__((ext_vector_type(8))) float v8f;
typedef __attribute__((ext_vector_type(4))) unsigned int v4u;
typedef __attribute__((ext_vector_type(8))) int v8i;
typedef __attribute__((ext_vector_type(4))) int v4i;

#define B_ 8
#define L_ 2048
#define D_ 1024
#define H_ 1024

enum { EPI_NONE = 0, EPI_BIAS, EPI_SUBC, EPI_ADDC, EPI_ROWSCALE, EPI_DECAY };

static constexpr int BM = 128, BN = 128, BK = 32;

// ---------------------------------------------------------------------------
// TDM: async-load one [128 rows x 32 floats] tile (row stride = strideElems)
// into LDS at byte offset ldsOff, padding +2 DWORDs after every 32 DWORDs so
// the LDS image has row stride BK+2 (matches As/Bs layout). D# per ISA ch.8.
// Toolchain uses the 6-arg builtin form: (g0 x4, g1 x8, g2 x4, g3 x4, x8, cpol)
// ---------------------------------------------------------------------------
__device__ __forceinline__ void tdm_tile_load(unsigned ldsOff, const float* g,
                                              long strideElems)
{
  unsigned long long ga = (unsigned long long)(uintptr_t)g;
  // group0: count=1 | lds_addr | global_addr(57b) | type=2
  v4u g0;
  g0.x = 1u;
  g0.y = ldsOff;
  g0.z = (unsigned)(ga & 0xffffffffu);
  g0.w = (unsigned)((ga >> 32) & 0x01ffffffu) | (2u << 30);
  // group1: data_size=4B, pad_enable, pad_interval=32dw, pad_amount=2dw,
  //         tensor dims huge (tiles always in-bounds), tile = 32 x 128.
  const unsigned dim0 = 0x7fffffffu, dim1 = 0x7fffffffu;
  unsigned long long s0 = (unsigned long long)strideElems;  // 48-bit, elems
  v8i g1;
  g1[0] = (int)((2u << 16) | (1u << 20) | (4u << 22) | (1u << 25));
  g1[1] = (int)((dim0 & 0xffffu) << 16);                       // dim0 lo16
  g1[2] = (int)(((dim0 >> 16) & 0xffffu) | ((dim1 & 0xffffu) << 16));
  g1[3] = (int)(((dim1 >> 16) & 0xffffu) | ((unsigned)BK << 16));  // tile_dim0
  g1[4] = (int)128u;                                           // tile_dim1
  g1[5] = (int)(unsigned)(s0 & 0xffffffffu);                   // stride0 lo32
  g1[6] = (int)(unsigned)((s0 >> 32) & 0xffffu);               // stride0 hi16
  g1[7] = 0;
  v4i z4 = {0, 0, 0, 0};                 // group2/group3: unused (2-D tile)
  v8i z8 = {0, 0, 0, 0, 0, 0, 0, 0};     // trailing group (unused)
  __builtin_amdgcn_tensor_load_to_lds(g0, g1, z4, z4, z8, 0);
}

// ---------------------------------------------------------------------------
// Generic 128x128x32 WMMA fp32 GEMM with fused epilogues.
//   TA : A stored [K, lda] accessed A[k][m]   (else A[m][k], lda = K-extent)
//   TB : B stored [N, ldb] accessed B[n][k]   (else B[k][n], ldb = N-extent)
//   KSC: scale A by kvec[k] while staging (W_next GEMM)
//   TDMA/TDMB: stage that operand with tensor_load_to_lds (requires the
//              operand to be K-contiguous in memory: !TA / TB), double-buffered
//              with a one-tile software pipeline on TENSORcnt.
// 256 threads = 8 wave32; wave grid 2(M) x 4(N); each wave: 4x2 tiles of 16x16.
// ---------------------------------------------------------------------------
template <bool TA, bool TB, int EPI, bool KSC, bool TDMA, bool TDMB>
__global__ __launch_bounds__(256) void gemm_wmma(
    const float* __restrict__ A, int lda, long sA,
    const float* __restrict__ Bm, int ldb, long sB,
    float* __restrict__ C, int ldc, long sC,
    int M, int N, int K,
    const float* __restrict__ bias,                       // EPI_BIAS
    const float* __restrict__ Cin, int ldcin, long sCin,  // EPI_SUBC / EPI_ADDC
    const float* __restrict__ csc,                        // EPI_ADDC per-batch scale
    const float* __restrict__ vec1,                       // lr (DECAY) / rowscale
    const float* __restrict__ vec2,                       // cwd (DECAY)
    int Lvec,
    const float* __restrict__ kvec)                       // KSC per-k A scale
{
  constexpr int NBA = TDMA ? 2 : 1;
  constexpr int NBB = TDMB ? 2 : 1;
  constexpr int NOPS = (TDMA ? 1 : 0) + (TDMB ? 1 : 0);  // TDM issues per tile

  const int bz = blockIdx.z;
  const int bm0 = blockIdx.y * BM;
  const int bn0 = blockIdx.x * BN;
  const long vecOff = (long)bz * Lvec;
  const float* Ab = A + (long)bz * sA;
  const float* Bb = Bm + (long)bz * sB;
  float* Cb = C + (long)bz * sC;

  if (EPI == EPI_DECAY) {
    // causal early-out: block entirely below diagonal (all j < i) -> zeros
    if (bn0 + BN <= bm0) {
      for (int t = threadIdx.x; t < BM * BN; t += 256) {
        int nl = t >> 7, ml = t & 127;
        Cb[(long)(bn0 + nl) * ldc + bm0 + ml] = 0.f;
      }
      return;
    }
  }

  __shared__ float As[NBA][BM][BK + 2];  // k-major rows, +2 pad (8B align kept)
  __shared__ float Bs[NBB][BN][BK + 2];

  const int lane = threadIdx.x & 31;
  const int wv = threadIdx.x >> 5;
  const int m0w = (wv & 1) * 64;   // wave M origin in block tile
  const int n0w = (wv >> 1) * 32;  // wave N origin
  const int fr = lane & 15;        // fragment row: m for A-frag, n for B-frag
  const int fk = (lane >> 4) * 2;  // fragment k pair base (0 or 2)

  v8f acc[4][2] = {};

  const int nt = K / BK;

  // TDM pipeline prologue: wave 0 kicks tile 0
  if ((TDMA || TDMB) && threadIdx.x < 32) {
    if (TDMA)
      tdm_tile_load((unsigned)(uintptr_t)&As[0][0][0],
                    Ab + (long)bm0 * lda, (long)lda);
    if (TDMB)
      tdm_tile_load((unsigned)(uintptr_t)&Bs[0][0][0],
                    Bb + (long)bn0 * ldb, (long)ldb);
  }

  for (int it = 0; it < nt; ++it) {
    const int k0 = it * BK;
    const int abuf = TDMA ? (it & 1) : 0;
    const int bbuf = TDMB ? (it & 1) : 0;

    // ------------- manual staging for non-TDM operands -------------
    if (!TDMA) {
      if (!TA) {
#pragma unroll
        for (int i = 0; i < 4; i++) {
          int lin = threadIdx.x + i * 256;  // float4 granules, 1024 total
          int r = lin >> 3, c4 = (lin & 7) * 4;
          const float4 v = *(const float4*)(Ab + (long)(bm0 + r) * lda + k0 + c4);
          As[0][r][c4 + 0] = v.x; As[0][r][c4 + 1] = v.y;
          As[0][r][c4 + 2] = v.z; As[0][r][c4 + 3] = v.w;
        }
      } else {
#pragma unroll
        for (int i = 0; i < 16; i++) {
          int lin = threadIdx.x + i * 256;   // 4096 scalars
          int ml = lin & 127, kl = lin >> 7; // coalesced along m
          float v = Ab[(long)(k0 + kl) * lda + bm0 + ml];
          if (KSC) v *= kvec[vecOff + k0 + kl];
          As[0][ml][kl] = v;
        }
      }
      int kn = k0 + BK;
      if (kn < K) {
        int lin = threadIdx.x * 16;
        if (!TA) { int r = lin >> 5, c = lin & 31;
          __builtin_prefetch(Ab + (long)(bm0 + r) * lda + kn + c, 0, 1);
        } else { int ml = lin & 127, kl = lin >> 7;
          __builtin_prefetch(Ab + (long)(kn + kl) * lda + bm0 + ml, 0, 1);
        }
      }
    }
    if (!TDMB) {
      if (TB) {
#pragma unroll
        for (int i = 0; i < 4; i++) {
          int lin = threadIdx.x + i * 256;
          int r = lin >> 3, c4 = (lin & 7) * 4;
          const float4 v = *(const float4*)(Bb + (long)(bn0 + r) * ldb + k0 + c4);
          Bs[0][r][c4 + 0] = v.x; Bs[0][r][c4 + 1] = v.y;
          Bs[0][r][c4 + 2] = v.z; Bs[0][r][c4 + 3] = v.w;
        }
      } else {
#pragma unroll
        for (int i = 0; i < 16; i++) {
          int lin = threadIdx.x + i * 256;
          int nl = lin & 127, kl = lin >> 7;
          Bs[0][nl][kl] = Bb[(long)(k0 + kl) * ldb + bn0 + nl];
        }
      }
      int kn = k0 + BK;
      if (kn < K) {
        int lin = threadIdx.x * 16;
        if (TB) { int r = lin >> 5, c = lin & 31;
          __builtin_prefetch(Bb + (long)(bn0 + r) * ldb + kn + c, 0, 1);
        } else { int nl = lin & 127, kl = lin >> 7;
          __builtin_prefetch(Bb + (long)(kn + kl) * ldb + bn0 + nl, 0, 1);
        }
      }
    }

    // ------------- TDM pipeline: issue tile it+1, retire tile it -------------
    if ((TDMA || TDMB) && threadIdx.x < 32) {
      if (it + 1 < nt) {
        const int kn = k0 + BK;
        const int nb = (it + 1) & 1;
        if (TDMA)
          tdm_tile_load((unsigned)(uintptr_t)&As[nb][0][0],
                        Ab + (long)bm0 * lda + kn, (long)lda);
        if (TDMB)
          tdm_tile_load((unsigned)(uintptr_t)&Bs[nb][0][0],
                        Bb + (long)bn0 * ldb + kn, (long)ldb);
        __builtin_amdgcn_s_wait_tensorcnt(NOPS);  // tile `it` landed in LDS
      } else {
        __builtin_amdgcn_s_wait_tensorcnt(0);
      }
    }
    __syncthreads();

    // ------------------- WMMA inner product -------------------
#pragma unroll
    for (int kk = 0; kk < BK; kk += 4) {
      v2f af[4], bf[2];
#pragma unroll
      for (int i = 0; i < 4; i++)
        af[i] = *(const v2f*)&As[abuf][m0w + i * 16 + fr][kk + fk];
#pragma unroll
      for (int j = 0; j < 2; j++)
        bf[j] = *(const v2f*)&Bs[bbuf][n0w + j * 16 + fr][kk + fk];
#pragma unroll
      for (int i = 0; i < 4; i++)
#pragma unroll
        for (int j = 0; j < 2; j++)
          acc[i][j] = __builtin_amdgcn_wmma_f32_16x16x4_f32(
              false, af[i], false, bf[j], (short)0, acc[i][j], false, false);
    }
    __syncthreads();
  }

  // ------------------- epilogue -------------------
  float cs = 1.f;
  if (EPI == EPI_ADDC && csc) cs = csc[bz];
  const int mlane = 8 * (lane >> 4);
#pragma unroll
  for (int i = 0; i < 4; i++) {
#pragma unroll
    for (int j = 0; j < 2; j++) {
#pragma unroll
      for (int r = 0; r < 8; r++) {
        int m = bm0 + m0w + i * 16 + r + mlane;  // C layout: vgpr r -> M=r / r+8
        int n = bn0 + n0w + j * 16 + fr;
        float v = acc[i][j][r];
        if (EPI == EPI_BIAS) v += bias[n];
        else if (EPI == EPI_SUBC) v -= Cin[(long)bz * sCin + (long)m * ldcin + n];
        else if (EPI == EPI_ADDC) v += cs * Cin[(long)bz * sCin + (long)m * ldcin + n];
        else if (EPI == EPI_ROWSCALE) v *= vec1[vecOff + m];
        if (EPI == EPI_DECAY) {
          // m = key index i, n = query index j; store transposed: T[j,i]
          float o = 0.f;
          if (n >= m)
            o = -vec1[vecOff + m] * __expf(vec2[vecOff + n] - vec2[vecOff + m]) * v;
          Cb[(long)n * ldc + m] = o;
        } else {
          Cb[(long)m * ldc + n] = v;
        }
      }
    }
  }
}

// ---------------------------------------------------------------------------
// lr / log_wd row-dot: one wave32 per (b,l) row of x.
// ---------------------------------------------------------------------------
__global__ __launch_bounds__(256) void rowdot_lrwd(
    const float* __restrict__ x, const float* __restrict__ w_lr,
    const float* __restrict__ b_lr, const float* __restrict__ lblr,
    const float* __restrict__ w_wd, const float* __restrict__ b_wd,
    const float* __restrict__ lbwd, float* __restrict__ lr,
    float* __restrict__ logwd, int D)
{
  int row = blockIdx.x * 8 + (threadIdx.x >> 5);
  int lane = threadIdx.x & 31;
  const float* xr = x + (long)row * D;
  float s1 = 0.f, s2 = 0.f;
  for (int d = lane; d < D; d += 32) {
    float xv = xr[d];
    s1 += xv * w_lr[d];
    s2 += xv * w_wd[d];
  }
#pragma unroll
  for (int off = 16; off > 0; off >>= 1) {
    s1 += __shfl_xor(s1, off, 32);
    s2 += __shfl_xor(s2, off, 32);
  }
  if (lane == 0) {
    float t1 = s1 + b_lr[0], t2 = s2 + b_wd[0];
    float sg1 = 1.f / (1.f + __expf(-t1));
    float sg2 = 1.f / (1.f + __expf(-t2));
    lr[row] = __expf(lblr[0]) * sg1;
    logwd[row] = log1pf(-__expf(lbwd[0]) * sg2);
  }
}

// ---------------------------------------------------------------------------
// Per-batch prefix sum of log_wd; derives wd_cross, gneg, wdl. One block/batch.
// ---------------------------------------------------------------------------
__global__ __launch_bounds__(256) void scan_decay(
    const float* __restrict__ logwd, const float* __restrict__ lr,
    float* __restrict__ cwd, float* __restrict__ wdcross,
    float* __restrict__ gneg, float* __restrict__ wdl, int L)
{
  __shared__ float s[L_];
  __shared__ float part[257];
  const int b = blockIdx.x, t = threadIdx.x;
  const int CH = L_ / 256;  // 8
  const float* w = logwd + (long)b * L;
  float run = 0.f, loc[8];
#pragma unroll
  for (int i = 0; i < CH; i++) { run += w[t * CH + i]; loc[i] = run; }
  part[t] = run;
  __syncthreads();
  if (t == 0) {
    float a = 0.f;
    for (int i = 0; i < 256; i++) { float p = part[i]; part[i] = a; a += p; }
    part[256] = a;
  }
  __syncthreads();
  float off = part[t];
#pragma unroll
  for (int i = 0; i < CH; i++) s[t * CH + i] = loc[i] + off;
  __syncthreads();
  float cL = s[L - 1];
#pragma unroll
  for (int i = 0; i < CH; i++) {
    int idx = t * CH + i;
    float c = s[idx];
    cwd[(long)b * L + idx] = c;
    wdcross[(long)b * L + idx] = __expf(c);
    gneg[(long)b * L + idx] = -lr[(long)b * L + idx] * __expf(cL - c);
  }
  if (t == 0) wdl[b] = __expf(cL);
}

// ---------------------------------------------------------------------------
extern "C" void kernel_launch(void* const* d_in, const int* in_sizes, int n_in,
                              void* d_out, int out_size, void* d_ws,
                              size_t ws_size, hipStream_t stream)
{
  (void)in_sizes; (void)n_in; (void)out_size; (void)ws_size;
  const float* x    = (const float*)d_in[0];
  const float* hid  = (const float*)d_in[1];
  const float* lblr = (const float*)d_in[2];
  const float* w_lr = (const float*)d_in[3];
  const float* b_lr = (const float*)d_in[4];
  const float* lbwd = (const float*)d_in[5];
  const float* w_wd = (const float*)d_in[6];
  const float* b_wd = (const float*)d_in[7];
  const float* w_q  = (const float*)d_in[8];
  const float* b_q  = (const float*)d_in[9];
  const float* w_k  = (const float*)d_in[10];
  const float* b_k  = (const float*)d_in[11];
  const float* w_v  = (const float*)d_in[12];
  const float* b_v  = (const float*)d_in[13];
  float* out = (float*)d_out;

  const size_t nLH = (size_t)B_ * L_ * H_;
  const size_t nLD = (size_t)B_ * L_ * D_;
  const size_t nLL = (size_t)B_ * L_ * L_;
  float* ws   = (float*)d_ws;
  float* Q    = ws;
  float* K    = Q + nLH;
  float* V    = K + nLH;       // becomes WpKV in place
  float* QWs  = V + nLD;
  float* T    = QWs + nLD;     // [B,L,L]
  float* lr   = T + nLL;
  float* lw   = lr + (size_t)B_ * L_;
  float* cwd  = lw + (size_t)B_ * L_;
  float* wdc  = cwd + (size_t)B_ * L_;
  float* gneg = wdc + (size_t)B_ * L_;
  float* wdl  = gneg + (size_t)B_ * L_;

  dim3 blk(256);

  // 1-2: scalar gates
  rowdot_lrwd<<<(B_ * L_) / 8, blk, 0, stream>>>(x, w_lr, b_lr, lblr, w_wd,
                                                 b_wd, lbwd, lr, lw, D_);
  scan_decay<<<B_, blk, 0, stream>>>(lw, lr, cwd, wdc, gneg, wdl, L_);

  // 3-5: projections  [L,D]@[D,H]+bias   (A via TDM)
  dim3 gQ(H_ / BN, L_ / BM, B_);
  gemm_wmma<false, false, EPI_BIAS, false, true, false><<<gQ, blk, 0, stream>>>(
      x, D_, (long)L_ * D_, w_q, H_, 0, Q, H_, (long)L_ * H_, L_, H_, D_,
      b_q, nullptr, 0, 0, nullptr, nullptr, nullptr, 0, nullptr);
  gemm_wmma<false, false, EPI_BIAS, false, true, false><<<gQ, blk, 0, stream>>>(
      x, D_, (long)L_ * D_, w_k, H_, 0, K, H_, (long)L_ * H_, L_, H_, D_,
      b_k, nullptr, 0, 0, nullptr, nullptr, nullptr, 0, nullptr);
  dim3 gV(D_ / BN, L_ / BM, B_);
  gemm_wmma<false, false, EPI_BIAS, false, true, false><<<gV, blk, 0, stream>>>(
      x, D_, (long)L_ * D_, w_v, D_, 0, V, D_, (long)L_ * D_, L_, D_, D_,
      b_v, nullptr, 0, 0, nullptr, nullptr, nullptr, 0, nullptr);

  // 6: WpKV = K@W_prevᵀ - V  (in place over V)   (A and B via TDM)
  gemm_wmma<false, true, EPI_SUBC, false, true, true><<<gV, blk, 0, stream>>>(
      K, H_, (long)L_ * H_, hid, H_, (long)D_ * H_, V, D_, (long)L_ * D_,
      L_, D_, H_, nullptr, V, D_, (long)L_ * D_, nullptr, nullptr, nullptr,
      0, nullptr);

  // 7: QWs = wd_cross[row] * (Q@W_prevᵀ)          (A and B via TDM)
  gemm_wmma<false, true, EPI_ROWSCALE, false, true, true><<<gV, blk, 0, stream>>>(
      Q, H_, (long)L_ * H_, hid, H_, (long)D_ * H_, QWs, D_, (long)L_ * D_,
      L_, D_, H_, nullptr, nullptr, 0, 0, nullptr, wdc, nullptr, L_, nullptr);

  // 8: T[j,i] = causal-decay of (K@Qᵀ)[i,j], stored transposed (A,B via TDM)
  dim3 gT(L_ / BN, L_ / BM, B_);
  gemm_wmma<false, true, EPI_DECAY, false, true, true><<<gT, blk, 0, stream>>>(
      K, H_, (long)L_ * H_, Q, H_, (long)L_ * H_, T, L_, (long)L_ * L_,
      L_, L_, H_, nullptr, nullptr, 0, 0, nullptr, lr, cwd, L_, nullptr);

  // 9: y = T@WpKV + QWs  -> d_out                 (A via TDM)
  gemm_wmma<false, false, EPI_ADDC, false, true, false><<<gV, blk, 0, stream>>>(
      T, L_, (long)L_ * L_, V, D_, (long)L_ * D_, out, D_, (long)L_ * D_,
      L_, D_, L_, nullptr, QWs, D_, (long)L_ * D_, nullptr, nullptr, nullptr,
      0, nullptr);

  // 10: W_next = (gneg⊙WpKV)ᵀ@K + wdl[b]*W_prev -> d_out + B*L*D (manual:
  //     A needs transpose + per-k scale during staging, TDM cannot do either)
  dim3 gW(H_ / BN, D_ / BM, B_);
  gemm_wmma<true, false, EPI_ADDC, true, false, false><<<gW, blk, 0, stream>>>(
      V, D_, (long)L_ * D_, K, H_, (long)L_ * H_, out + nLD, H_,
      (long)D_ * H_, D_, H_, L_, nullptr, hid, H_, (long)D_ * H_, wdl,
      nullptr, nullptr, L_, gneg);
}